// GNN_2946347565789
// MI455X (gfx1250) — compile-verified
//
#include <hip/hip_runtime.h>
#include <hip/hip_bf16.h>

typedef __attribute__((ext_vector_type(2))) float v2f;
typedef __attribute__((ext_vector_type(8))) float v8f;

#define D 32  // feature dim (both layers), fixed by the reference

// ---------------------------------------------------------------------------
// Zero a float buffer (agg/deg scratch must be re-zeroed every call; the
// harness poisons ws once and never restores it).
// ---------------------------------------------------------------------------
__global__ void sage_zero_kernel(float* __restrict__ p, long long n) {
    long long i = (long long)blockIdx.x * blockDim.x + threadIdx.x;
    if (i < n) p[i] = 0.0f;
}

// ---------------------------------------------------------------------------
// Edge scatter-add: one lane per (edge, feature). 32 consecutive lanes handle
// the 32 features of one edge -> coalesced atomic bursts into L2 (x, agg and
// deg all fit in the 192MB L2, so these atomics never touch HBM in steady
// state). Degree is accumulated only when add_deg != 0 (layer 1); layer 2
// reuses it since it depends only on dst.
// ---------------------------------------------------------------------------
__global__ void sage_scatter_kernel(const float* __restrict__ xsrc,
                                    const long long* __restrict__ src,
                                    const long long* __restrict__ dst,
                                    float* __restrict__ agg,
                                    float* __restrict__ deg,
                                    int add_deg,
                                    long long E) {
    long long t = (long long)blockIdx.x * blockDim.x + threadIdx.x;
    long long e = t >> 5;
    int f = (int)(t & 31);
    if (e >= E) return;
    long long s = src[e];
    long long d = dst[e];
    atomicAdd(&agg[d * D + f], xsrc[s * D + f]);
    if (add_deg && f == 0) atomicAdd(&deg[d], 1.0f);
}

// ---------------------------------------------------------------------------
// Fused SAGE update: out[i,:] = relu( (agg[i,:]/max(deg[i],1)) @ Wl + b
//                                     + xroot[i,:] @ Wr )
// One wave per 16-node tile, V_WMMA_F32_16X16X4_F32 (fp32 in/out -> matches
// the fp32 reference numerically). 8 K-chunks of 4, 2 N-tiles of 16,
// 2 matrices -> 32 WMMAs per tile, A fragments reused across both N-tiles.
// Weights + bias staged in LDS.
//
// WMMA fp32 fragment layouts (ISA 7.12.2):
//   A (16x4):  lane = M row (lanes 16-31 mirror M with K pair {2,3});
//              v2f = {A[M][4k+koff], A[M][4k+koff+1]}, koff = 2*(lane>=16)
//   B (4x16):  lane = N col (same half split selects K pair)
//   C/D 16x16: VGPR v, lane l -> M = v + 8*(l>=16), N = l&15
// ---------------------------------------------------------------------------
__global__ void __launch_bounds__(256)
sage_wmma_kernel(const float* __restrict__ agg,
                 const float* __restrict__ xroot,
                 const float* __restrict__ deg,
                 const float* __restrict__ Wl,
                 const float* __restrict__ Wr,
                 const float* __restrict__ bias,
                 float* __restrict__ out,
                 int ntiles, int nnodes) {
    __shared__ float sWl[D * D];
    __shared__ float sWr[D * D];
    __shared__ float sb[D];

    for (int i = threadIdx.x; i < D * D; i += blockDim.x) {
        sWl[i] = Wl[i];
        sWr[i] = Wr[i];
    }
    if (threadIdx.x < D) sb[threadIdx.x] = bias[threadIdx.x];
    __syncthreads();

    const int wave = threadIdx.x >> 5;
    const int lane = threadIdx.x & 31;
    const int tile = blockIdx.x * (blockDim.x >> 5) + wave;
    if (tile >= ntiles) return;  // whole wave exits together (EXEC stays all-1)

    const int half = lane >> 4;   // 0: K{0,1}, 1: K{2,3}
    const int koff = half * 2;
    const int col  = lane & 15;

    // Clamp row for loads so every lane stays active (WMMA needs EXEC==all 1s);
    // stores are predicated instead. N=100000 is a multiple of 16, so the
    // clamp is a no-op here but keeps the kernel shape-safe.
    int row = tile * 16 + (lane & 15);
    int rowL = row < nnodes ? row : nnodes - 1;

    const float invd = 1.0f / fmaxf(deg[rowL], 1.0f);

    const v2f* aggv = (const v2f*)agg;
    const v2f* xv   = (const v2f*)xroot;

    v2f Az[8], Ax[8];
#pragma unroll
    for (int k = 0; k < 8; ++k) {
        int idx = rowL * D + 4 * k + koff;   // even -> 8B aligned
        v2f a = aggv[idx >> 1];
        Az[k] = a * invd;                    // fuse mean-aggregation divide
        Ax[k] = xv[idx >> 1];
    }

#pragma unroll
    for (int n = 0; n < 2; ++n) {
        const float bb = sb[n * 16 + col];   // bias depends only on N column
        v8f c = {bb, bb, bb, bb, bb, bb, bb, bb};

#pragma unroll
        for (int k = 0; k < 8; ++k) {
            v2f b;
            b.x = sWl[(4 * k + koff) * D + n * 16 + col];
            b.y = sWl[(4 * k + koff + 1) * D + n * 16 + col];
            c = __builtin_amdgcn_wmma_f32_16x16x4_f32(
                    false, Az[k], false, b, (short)0, c, false, false);
        }
#pragma unroll
        for (int k = 0; k < 8; ++k) {
            v2f b;
            b.x = sWr[(4 * k + koff) * D + n * 16 + col];
            b.y = sWr[(4 * k + koff + 1) * D + n * 16 + col];
            c = __builtin_amdgcn_wmma_f32_16x16x4_f32(
                    false, Ax[k], false, b, (short)0, c, false, false);
        }

#pragma unroll
        for (int v = 0; v < 8; ++v) {
            int m = tile * 16 + v + half * 8;
            if (m < nnodes)
                out[m * D + n * 16 + col] = fmaxf(c[v], 0.0f);
        }
    }
}

// ---------------------------------------------------------------------------
// Pipeline:
//   L1: zero(agg,deg) -> scatter(x, +deg) -> wmma(agg,x,W1) -> d_out
//   L2: zero(agg)     -> scatter(d_out)   -> wmma(agg,d_out,W2) -> d_out
// Layer-1 output lives in d_out (h1); layer-2 GEMM reads only its own 16 rows
// of d_out before overwriting them, so in-place is safe. ws holds agg (N*32
// floats) + deg (N floats) = ~13.2 MB.
// ---------------------------------------------------------------------------
extern "C" void kernel_launch(void* const* d_in, const int* in_sizes, int n_in,
                              void* d_out, int out_size, void* d_ws, size_t ws_size,
                              hipStream_t stream) {
    const float*     x   = (const float*)d_in[0];
    const long long* ei  = (const long long*)d_in[1];   // int64 edge_index [2,E]
    const float*     W1l = (const float*)d_in[2];
    const float*     W1r = (const float*)d_in[3];
    const float*     b1  = (const float*)d_in[4];
    const float*     W2l = (const float*)d_in[5];
    const float*     W2r = (const float*)d_in[6];
    const float*     b2  = (const float*)d_in[7];
    float* out = (float*)d_out;

    const long long N = (long long)in_sizes[0] / D;   // 100000
    const long long E = (long long)in_sizes[1] / 2;   // 1600000

    float* agg = (float*)d_ws;
    float* deg = agg + N * D;
    const long long* src = ei;
    const long long* dst = ei + E;

    const int nnodes = (int)N;
    const int ntiles = (nnodes + 15) / 16;
    const int gemm_blocks = (ntiles + 7) / 8;       // 8 waves/block (wave32)
    const long long scat_t = E * 32;
    const int scat_blocks = (int)((scat_t + 255) / 256);

    // ----- layer 1 -----
    {
        long long z = N * D + N;  // agg + deg
        sage_zero_kernel<<<(int)((z + 255) / 256), 256, 0, stream>>>(agg, z);
    }
    sage_scatter_kernel<<<scat_blocks, 256, 0, stream>>>(x, src, dst, agg, deg, 1, E);
    sage_wmma_kernel<<<gemm_blocks, 256, 0, stream>>>(agg, x, deg, W1l, W1r, b1,
                                                      out, ntiles, nnodes);

    // ----- layer 2 (deg reused) -----
    {
        long long z = N * D;      // agg only
        sage_zero_kernel<<<(int)((z + 255) / 256), 256, 0, stream>>>(agg, z);
    }
    sage_scatter_kernel<<<scat_blocks, 256, 0, stream>>>(out, src, dst, agg, deg, 0, E);
    sage_wmma_kernel<<<gemm_blocks, 256, 0, stream>>>(agg, out, deg, W2l, W2r, b2,
                                                      out, ntiles, nnodes);
}